// AttnDecoder_20933670601086
// MI455X (gfx1250) — compile-verified
//
#include <hip/hip_runtime.h>
#include <math.h>

#define H 1024
#define V 50257
#define L 2048

typedef __attribute__((ext_vector_type(2))) float v2f;
typedef __attribute__((ext_vector_type(4))) float v4f;
typedef __attribute__((ext_vector_type(8))) float v8f;

// ---------------------------------------------------------------------------
// Generic mat-vec  y[r] = act( sum_c W[r][c]*x[c] + bias[r] )
// using V_WMMA_F32_16X16X4_F32.  One wave -> 16 output rows.
//
// K is processed 8 columns per iteration with ONE global_load_b128 per lane:
//   lane l<16  loads W[row][k..k+3]   (tile K-slots 0,1 use dwords 0,1)
//   lane 16+l  loads W[row][k+4..k+7] (tile K-slots 2,3 use dwords 0,1)
//   wmma #1 covers columns {k,k+1,k+4,k+5}   (dwords 0-1)
//   wmma #2 covers columns {k+2,k+3,k+6,k+7} (dwords 2-3)
// (dot products are K-order agnostic, so non-contiguous tile columns are fine)
// B carries x in column N=0 only: lanes 0 and 16 (same koff as their W load).
// D column N=0: lane 0 = rows 0..7 in acc[0..7], lane 16 = rows 8..15.
// Requires C % 8 == 0 (true for all uses: 1024 / 2048).
// ---------------------------------------------------------------------------
template <int ACT>
__global__ __launch_bounds__(256)
void k_matvec_wmma(const float* __restrict__ W, const float* __restrict__ x,
                   const float* __restrict__ bias, float* __restrict__ y,
                   int R, int C) {
  const int lane = threadIdx.x & 31;
  const int wave = threadIdx.x >> 5;
  const int rowbase = (blockIdx.x * 8 + wave) << 4;     // 16 rows per wave
  const int rl   = lane & 15;
  const int koff = (lane >> 4) << 2;                    // 0 or 4
  int row = rowbase + rl;
  int rclamp = row < R ? row : (R - 1);                 // keep loads in-bounds
  const float* wrow = W + (size_t)rclamp * (size_t)C;

  const bool isCol0 = (rl == 0);                        // lanes 0 and 16
  v8f acc = {};
  for (int k = 0; k < C; k += 8) {
    v4f a4 = *(const v4f*)(wrow + k + koff);            // global_load_b128
    v4f x4 = *(const v4f*)(x + k + koff);               // broadcast-ish, L1/L2 hit
    v2f a0 = {a4.x, a4.y};
    v2f a1 = {a4.z, a4.w};
    v2f b0, b1;
    b0.x = isCol0 ? x4.x : 0.0f;
    b0.y = isCol0 ? x4.y : 0.0f;
    b1.x = isCol0 ? x4.z : 0.0f;
    b1.y = isCol0 ? x4.w : 0.0f;
    acc = __builtin_amdgcn_wmma_f32_16x16x4_f32(
        false, a0, false, b0, (short)0, acc, false, false);
    acc = __builtin_amdgcn_wmma_f32_16x16x4_f32(
        false, a1, false, b1, (short)0, acc, false, false);
  }
  if (rl == 0) {
    int base = rowbase + ((lane >> 4) << 3);            // lane0 -> +0, lane16 -> +8
#pragma unroll
    for (int m = 0; m < 8; ++m) {
      int r = base + m;
      if (r < R) {
        float v = acc[m] + bias[r];
        if (ACT == 1) v = fmaxf(v, 0.0f);
        y[r] = v;
      }
    }
  }
}

// ---------------------------------------------------------------------------
// Embedding gather + build cat = [emb, h0] and the emb half of cat2
// ---------------------------------------------------------------------------
__global__ void k_prep(const int* __restrict__ inp,
                       const float* __restrict__ hidden,
                       const float* __restrict__ emb_table,
                       float* __restrict__ cat, float* __restrict__ cat2) {
  int t = blockIdx.x * blockDim.x + threadIdx.x;        // 0..2047
  int idx = inp[0];
  if (t < H) {
    float e = emb_table[(size_t)idx * H + t];
    cat[t]  = e;
    cat2[t] = e;
  } else if (t < 2 * H) {
    cat[t] = hidden[t - H];
  }
}

// ---------------------------------------------------------------------------
// Softmax over L=2048 (single workgroup, deterministic block reduction)
// ---------------------------------------------------------------------------
__global__ void k_softmax(const float* __restrict__ logits,
                          float* __restrict__ w, float* __restrict__ out_copy) {
  __shared__ float red[1024];
  int t = threadIdx.x;
  float m = -3.402823466e+38f;
  for (int i = t; i < L; i += 1024) m = fmaxf(m, logits[i]);
  red[t] = m; __syncthreads();
  for (int s = 512; s > 0; s >>= 1) {
    if (t < s) red[t] = fmaxf(red[t], red[t + s]);
    __syncthreads();
  }
  m = red[0]; __syncthreads();
  float sum = 0.0f;
  for (int i = t; i < L; i += 1024) sum += expf(logits[i] - m);
  red[t] = sum; __syncthreads();
  for (int s = 512; s > 0; s >>= 1) {
    if (t < s) red[t] += red[t + s];
    __syncthreads();
  }
  float inv = 1.0f / red[0];
  for (int i = t; i < L; i += 1024) {
    float wv = expf(logits[i] - m) * inv;
    w[i] = wv;
    out_copy[i] = wv;                                   // third output
  }
}

// ---------------------------------------------------------------------------
// attn_applied[h] = sum_l w[l] * enc[l][h]   (coalesced across h)
// ---------------------------------------------------------------------------
__global__ void k_attn_apply(const float* __restrict__ w,
                             const float* __restrict__ enc,
                             float* __restrict__ dst) {
  int h = blockIdx.x * blockDim.x + threadIdx.x;        // 0..1023
  float s = 0.0f;
  for (int l = 0; l < L; ++l) s += w[l] * enc[(size_t)l * H + h];
  dst[h] = s;
}

// ---------------------------------------------------------------------------
// GRU gates (PyTorch cell), writes h_new to ws and to d_out[V..V+H)
// ---------------------------------------------------------------------------
__global__ void k_gru_gate(const float* __restrict__ gi,
                           const float* __restrict__ gh,
                           const float* __restrict__ h0,
                           float* __restrict__ hnew, float* __restrict__ out_h) {
  int i = blockIdx.x * blockDim.x + threadIdx.x;        // 0..1023
  float r = 1.0f / (1.0f + expf(-(gi[i] + gh[i])));
  float z = 1.0f / (1.0f + expf(-(gi[H + i] + gh[H + i])));
  float n = tanhf(gi[2 * H + i] + r * gh[2 * H + i]);
  float h = (1.0f - z) * n + z * h0[i];
  hnew[i]  = h;
  out_h[i] = h;                                         // second output
}

// ---------------------------------------------------------------------------
// log_softmax over V=50257 (single workgroup, 3 passes over ~200KB in L2)
// ---------------------------------------------------------------------------
__global__ void k_logsoftmax(const float* __restrict__ logits,
                             float* __restrict__ out) {
  __shared__ float red[1024];
  int t = threadIdx.x;
  float m = -3.402823466e+38f;
  for (int i = t; i < V; i += 1024) m = fmaxf(m, logits[i]);
  red[t] = m; __syncthreads();
  for (int s = 512; s > 0; s >>= 1) {
    if (t < s) red[t] = fmaxf(red[t], red[t + s]);
    __syncthreads();
  }
  m = red[0]; __syncthreads();
  float sum = 0.0f;
  for (int i = t; i < V; i += 1024) sum += expf(logits[i] - m);
  red[t] = sum; __syncthreads();
  for (int s = 512; s > 0; s >>= 1) {
    if (t < s) red[t] += red[t + s];
    __syncthreads();
  }
  float lse = m + logf(red[0]);
  for (int i = t; i < V; i += 1024) out[i] = logits[i] - lse;
}

// ---------------------------------------------------------------------------
extern "C" void kernel_launch(void* const* d_in, const int* in_sizes, int n_in,
                              void* d_out, int out_size, void* d_ws, size_t ws_size,
                              hipStream_t stream) {
  const int*   inp       = (const int*)  d_in[0];
  const float* hidden    = (const float*)d_in[1];   // [1,1,H]
  const float* enc       = (const float*)d_in[2];   // [L,H]
  const float* emb_table = (const float*)d_in[3];   // [V,H]
  const float* attn_W    = (const float*)d_in[4];   // [L,2H]
  const float* attn_b    = (const float*)d_in[5];
  const float* comb_W    = (const float*)d_in[6];   // [H,2H]
  const float* comb_b    = (const float*)d_in[7];
  const float* gru_W_ih  = (const float*)d_in[8];   // [3H,H]
  const float* gru_W_hh  = (const float*)d_in[9];   // [3H,H]
  const float* gru_b_ih  = (const float*)d_in[10];
  const float* gru_b_hh  = (const float*)d_in[11];
  const float* out_W     = (const float*)d_in[12];  // [V,H]
  const float* out_b     = (const float*)d_in[13];

  float* out = (float*)d_out;   // [logp V | h_new H | attn_w L]

  float* ws     = (float*)d_ws;
  float* cat    = ws;            // 2H
  float* cat2   = ws + 2048;     // 2H
  float* logits = ws + 4096;     // L
  float* attnw  = ws + 6144;     // L
  float* xbuf   = ws + 8192;     // H
  float* gi     = ws + 9216;     // 3H
  float* gh     = ws + 12288;    // 3H
  float* hnew   = ws + 15360;    // H
  float* olog   = ws + 16384;    // V

  // 1) embed + cat
  k_prep<<<8, 256, 0, stream>>>(inp, hidden, emb_table, cat, cat2);
  // 2) attention logits: [2048 x 2048] @ cat
  k_matvec_wmma<0><<<L / 128, 256, 0, stream>>>(attn_W, cat, attn_b, logits, L, 2 * H);
  // 3) softmax -> attn weights (also 3rd output)
  k_softmax<<<1, 1024, 0, stream>>>(logits, attnw, out + V + H);
  // 4) attn_applied = attnw @ encoder_out  -> second half of cat2
  k_attn_apply<<<H / 256, 256, 0, stream>>>(attnw, enc, cat2 + H);
  // 5) x = relu(comb_W @ cat2 + b)
  k_matvec_wmma<1><<<H / 128, 256, 0, stream>>>(comb_W, cat2, comb_b, xbuf, H, 2 * H);
  // 6) GRU input / hidden projections
  k_matvec_wmma<0><<<(3 * H) / 128, 256, 0, stream>>>(gru_W_ih, xbuf, gru_b_ih, gi, 3 * H, H);
  k_matvec_wmma<0><<<(3 * H) / 128, 256, 0, stream>>>(gru_W_hh, hidden, gru_b_hh, gh, 3 * H, H);
  // 7) gates -> h_new (also 2nd output)
  k_gru_gate<<<H / 256, 256, 0, stream>>>(gi, gh, hidden, hnew, out + V);
  // 8) output logits: [50257 x 1024] @ h_new  (dominant 201MB stream)
  k_matvec_wmma<0><<<(V + 127) / 128, 256, 0, stream>>>(out_W, hnew, out_b, olog, V, H);
  // 9) log_softmax -> logp (1st output)
  k_logsoftmax<<<1, 1024, 0, stream>>>(olog, out);
}